// HiggsAudioTokenizerResidualVectorQuantization_53584011984913
// MI455X (gfx1250) — compile-verified
//
#include <hip/hip_runtime.h>

typedef float v2f __attribute__((ext_vector_type(2)));
typedef float v8f __attribute__((ext_vector_type(8)));
typedef unsigned int u32x4 __attribute__((ext_vector_type(4)));
typedef int i32x4 __attribute__((ext_vector_type(4)));
typedef int i32x8 __attribute__((ext_vector_type(8)));

#define B_   8
#define DIM_ 1024
#define T_   2048
#define CD_  128
#define K_   1024
#define Q_   8
#define M_   (B_ * T_)   // 16384 rows

// TDM D# group1 word0: data_size=4B (2<<16), pad_enable (1<<20),
// pad_interval=128 DWORDs (6<<22), pad_amount=4 DWORDs (3<<25)
// -> LDS row stride 132 floats, matching the padded shared arrays.
#define TDM_PADCFG ((2u << 16) | (1u << 20) | (6u << 22) | (3u << 25))

__device__ __forceinline__ unsigned rfl(unsigned x) {
  return (unsigned)__builtin_amdgcn_readfirstlane((int)x);
}

__device__ __forceinline__ void tdm_call(u32x4 g0, i32x8 g1, i32x4 g2, i32x4 g3) {
#if defined(__clang_major__) && (__clang_major__ >= 23)
  i32x8 g4 = (i32x8){};
  __builtin_amdgcn_tensor_load_to_lds(g0, g1, g2, g3, g4, 0);
#else
  __builtin_amdgcn_tensor_load_to_lds(g0, g1, g2, g3, 0);
#endif
}

// Generic TDM load: tensor_dim0 x tensor_dim1 tensor, tile0 x tile1 tile,
// contiguous-dim stride0 (elements), optional gather via mode_bits/g2/g3.
__device__ __forceinline__ void tdm_load(unsigned ldsoff, const void* gptr,
                                         unsigned td0, unsigned td1,
                                         unsigned tile0, unsigned tile1,
                                         unsigned stride0,
                                         i32x4 g2, i32x4 g3,
                                         unsigned mode_bits) {
  unsigned long long ga = (unsigned long long)(size_t)gptr;
  u32x4 g0;
  g0.x = rfl(1u | mode_bits);                                  // count=1 (+gather)
  g0.y = rfl(ldsoff);                                          // lds_addr
  g0.z = rfl((unsigned)(ga & 0xFFFFFFFFull));                  // global_addr lo
  g0.w = rfl(((unsigned)(ga >> 32) & 0x1FFFFFFu) | (2u << 30)); // addr hi | type=2
  i32x8 g1;
  g1[0] = (int)rfl(TDM_PADCFG);                                // mask=0,dsize,pad
  g1[1] = (int)rfl((td0 & 0xFFFFu) << 16);                     // tensor_dim0 lo16
  g1[2] = (int)rfl(((td0 >> 16) & 0xFFFFu) | ((td1 & 0xFFFFu) << 16));
  g1[3] = (int)rfl(((td1 >> 16) & 0xFFFFu) | ((tile0 & 0xFFFFu) << 16));
  g1[4] = (int)rfl(tile1 & 0xFFFFu);                           // tile_dim1 (or #idx)
  g1[5] = (int)rfl(stride0);                                   // dim0 stride lo32
  g1[6] = 0;
  g1[7] = 0;
  tdm_call(g0, g1, g2, g3);
}

// -------------------------------------------------------------------------
// enormAdj[q][k] = 0.5*||E_qk||^2 - b_in_q . E_qk   (folds bias out of GEMMs)
// -------------------------------------------------------------------------
__global__ void rvq_enorm(const float* __restrict__ embed,
                          const float* __restrict__ b_in,
                          float* __restrict__ enormAdj) {
  int gid = blockIdx.x * blockDim.x + threadIdx.x;
  if (gid >= Q_ * K_) return;
  int q = gid / K_;
  const float* E  = embed + (size_t)gid * CD_;
  const float* bi = b_in + (size_t)q * CD_;
  float s = 0.f;
  for (int c = 0; c < CD_; ++c) {
    float e = E[c];
    s += 0.5f * e * e - bi[c] * e;
  }
  enormAdj[gid] = s;
}

// -------------------------------------------------------------------------
// Assign: P = R^T * Wi (16x128 per wave, WMMA f32), then scores P.E^T vs all
// K=1024 codes. E tiles streamed into double-buffered LDS by the TDM
// (tensor_load_to_lds, hardware-padded to stride 132), overlapped with the
// WMMA score loop. argmin -> idx[m].
// Block = 128 threads = 4 waves; each wave owns 16 consecutive t rows.
// -------------------------------------------------------------------------
__global__ void __launch_bounds__(128)
rvq_assign(const float* __restrict__ R, const float* __restrict__ Wi,
           const float* __restrict__ E, const float* __restrict__ enormAdj,
           int* __restrict__ idx) {
  __shared__ float h_lds[4][16][132];   // per-wave P tile, padded stride
  __shared__ float e_lds[2][16][132];   // TDM double-buffered E tiles
  __shared__ float redd[4][16][16];
  __shared__ int   redi[4][16][16];

  const int tid  = threadIdx.x;
  const int wave = tid >> 5;
  const int lane = tid & 31;
  const int n    = lane & 15;        // M (A) or N (B/C/D) index
  const int hi   = lane >> 4;        // half-wave select
  const int kb   = hi * 2;           // K-pair base for 16x16x4 layout

  const int tile = blockIdx.x * 4 + wave;
  const int m0   = tile * 16;
  const int b    = m0 / T_;
  const int tb   = m0 % T_;
  const float* Rb = R + (size_t)b * DIM_ * T_ + tb;

  const unsigned ebase    = (unsigned)(size_t)&e_lds[0][0][0];
  const unsigned ebufsize = 16u * 132u * 4u;
  const i32x4 z4 = (i32x4){};

  // Kick off TDM for E chunk 0 while we do phase 1.
  if (wave == 0)
    tdm_load(ebase, E, /*td0*/ 16 * CD_, /*td1*/ 1,
             /*tile0*/ 16 * CD_, /*tile1*/ 1, /*stride0*/ 16 * CD_, z4, z4, 0);

  // ---- Phase 1: P = R^T * Wi, 16x128 in 8 accumulators ----
  v8f hc[8];
  #pragma unroll
  for (int i = 0; i < 8; ++i) hc[i] = (v8f){};

  for (int d0 = 0; d0 < DIM_; d0 += 4) {
    v2f a;
    a.x = Rb[(size_t)(d0 + kb)     * T_ + n];   // lanes coalesced over t
    a.y = Rb[(size_t)(d0 + kb + 1) * T_ + n];
    #pragma unroll
    for (int nc = 0; nc < 8; ++nc) {
      v2f bm;
      bm.x = Wi[(size_t)(d0 + kb)     * CD_ + nc * 16 + n]; // coalesced over c
      bm.y = Wi[(size_t)(d0 + kb + 1) * CD_ + nc * 16 + n];
      hc[nc] = __builtin_amdgcn_wmma_f32_16x16x4_f32(
          false, a, false, bm, (short)0, hc[nc], false, false);
    }
  }
  // spill P to LDS (C/D layout: row j+8*hi, col nc*16+n)
  #pragma unroll
  for (int nc = 0; nc < 8; ++nc)
    #pragma unroll
    for (int j = 0; j < 8; ++j)
      h_lds[wave][j + 8 * hi][nc * 16 + n] = hc[nc][j];

  float bestd[8];
  int   bestk[8];
  #pragma unroll
  for (int j = 0; j < 8; ++j) { bestd[j] = 3.4e38f; bestk[j] = 0; }

  // ---- Phase 2: scores against all codes, 16 codes per chunk ----
  for (int kc = 0; kc < K_ / 16; ++kc) {
    const int buf = kc & 1;
    if (wave == 0) {
      if (kc + 1 < K_ / 16) {
        // prefetch next chunk into the other buffer, then wait for current
        tdm_load(ebase + (unsigned)((kc + 1) & 1) * ebufsize,
                 E + (size_t)(kc + 1) * 16 * CD_,
                 16 * CD_, 1, 16 * CD_, 1, 16 * CD_, z4, z4, 0);
        __builtin_amdgcn_s_wait_tensorcnt(1);
      } else {
        __builtin_amdgcn_s_wait_tensorcnt(0);
      }
    }
    __syncthreads();   // chunk kc visible to all waves

    v8f s = (v8f){};
    for (int c0 = 0; c0 < CD_; c0 += 4) {
      v2f a, bm;
      a.x  = h_lds[wave][n][c0 + kb];
      a.y  = h_lds[wave][n][c0 + kb + 1];
      bm.x = e_lds[buf][n][c0 + kb];
      bm.y = e_lds[buf][n][c0 + kb + 1];
      s = __builtin_amdgcn_wmma_f32_16x16x4_f32(
          false, a, false, bm, (short)0, s, false, false);
    }

    int   code = kc * 16 + n;
    float ea   = enormAdj[code];
    #pragma unroll
    for (int j = 0; j < 8; ++j) {
      float dist = ea - s[j];                  // row m = j + 8*hi
      if (dist < bestd[j]) { bestd[j] = dist; bestk[j] = code; }
    }
    __syncthreads();   // all waves done with buf before TDM reuses it
  }

  // ---- cross-lane argmin reduction (16 candidate lanes per row) ----
  #pragma unroll
  for (int j = 0; j < 8; ++j) {
    redd[wave][j + 8 * hi][n] = bestd[j];
    redi[wave][j + 8 * hi][n] = bestk[j];
  }
  __syncthreads();
  if (lane < 16) {
    float bd = redd[wave][lane][0];
    int   bk = redi[wave][lane][0];
    for (int c = 1; c < 16; ++c) {
      float d2 = redd[wave][lane][c];
      int   k2 = redi[wave][lane][c];
      if (d2 < bd || (d2 == bd && k2 < bk)) { bd = d2; bk = k2; }
    }
    idx[m0 + lane] = bk;
  }
}

// -------------------------------------------------------------------------
// Decode: q = E[idx] * Wo + bo ; R -= q.  Codebook rows gathered by the TDM
// in gather mode (16-bit row indices in D# groups 2/3) into wave-private LDS.
// Output tile transposed through LDS so [B,DIM,T] stores are t-contiguous.
// -------------------------------------------------------------------------
__global__ void __launch_bounds__(128)
rvq_decode(float* __restrict__ R, const float* __restrict__ E,
           const float* __restrict__ Wo, const float* __restrict__ bo,
           const int* __restrict__ idx) {
  __shared__ float a_lds[4][16][132];   // gathered codebook rows per wave
  __shared__ float tq[4][16][17];       // 16x16 output tile transpose buffer

  const int tid  = threadIdx.x;
  const int wave = tid >> 5;
  const int lane = tid & 31;
  const int n    = lane & 15;
  const int hi   = lane >> 4;
  const int kb   = hi * 2;

  const int tile = blockIdx.x * 4 + wave;
  const int m0   = tile * 16;
  const int b    = m0 / T_;
  const int tb   = m0 % T_;
  float* Rb = R + (size_t)b * DIM_ * T_ + tb;

  // gather A rows via TDM gather mode: a_lds[wave][m][c] = E[idx[m0+m]][c]
  unsigned codes[16];
  #pragma unroll
  for (int rr = 0; rr < 16; ++rr)
    codes[rr] = (unsigned)idx[m0 + rr] & 0xFFFFu;
  i32x4 g2, g3;
  #pragma unroll
  for (int w = 0; w < 4; ++w) {
    g2[w] = (int)rfl(codes[2 * w]     | (codes[2 * w + 1] << 16));
    g3[w] = (int)rfl(codes[2 * w + 8] | (codes[2 * w + 9] << 16));
  }
  const unsigned aoff =
      (unsigned)(size_t)&a_lds[0][0][0] + (unsigned)wave * 16u * 132u * 4u;
  tdm_load(aoff, E, /*td0*/ CD_, /*td1*/ K_, /*tile0*/ CD_,
           /*tile1(#idx)*/ 16, /*stride0*/ CD_, g2, g3, /*gather*/ 0x80000000u);
  __builtin_amdgcn_s_wait_tensorcnt(0);   // wave-private slice: no barrier needed

  for (int n0 = 0; n0 < DIM_; n0 += 16) {
    v8f s = (v8f){};
    for (int c0 = 0; c0 < CD_; c0 += 4) {
      v2f a, bm;
      a.x  = a_lds[wave][n][c0 + kb];
      a.y  = a_lds[wave][n][c0 + kb + 1];
      bm.x = Wo[(size_t)(c0 + kb)     * DIM_ + n0 + n];  // coalesced over d
      bm.y = Wo[(size_t)(c0 + kb + 1) * DIM_ + n0 + n];
      s = __builtin_amdgcn_wmma_f32_16x16x4_f32(
          false, a, false, bm, (short)0, s, false, false);
    }
    float bias = bo[n0 + n];
    #pragma unroll
    for (int j = 0; j < 8; ++j)
      tq[wave][j + 8 * hi][n] = s[j] + bias;   // [t-row][d-col]
    __syncthreads();

    // store transposed: 32 lanes = 2 d-columns x 16 consecutive t
    #pragma unroll
    for (int dd = 0; dd < 16; dd += 2) {
      int d = n0 + dd + hi;
      float v = tq[wave][n][dd + hi];
      size_t off = (size_t)d * T_ + n;
      Rb[off] = Rb[off] - v;                   // residual -= q (coalesced)
    }
    __syncthreads();
  }
}

// -------------------------------------------------------------------------
// out = embeddings - residual_final   (acc == emb - R identity)
// -------------------------------------------------------------------------
__global__ void rvq_final(const float4* __restrict__ emb,
                          const float4* __restrict__ R,
                          float4* __restrict__ out, int n4) {
  for (int i = blockIdx.x * blockDim.x + threadIdx.x; i < n4;
       i += gridDim.x * blockDim.x) {
    float4 e = emb[i], r = R[i];
    float4 o;
    o.x = e.x - r.x; o.y = e.y - r.y; o.z = e.z - r.z; o.w = e.w - r.w;
    out[i] = o;
  }
}

// -------------------------------------------------------------------------
extern "C" void kernel_launch(void* const* d_in, const int* in_sizes, int n_in,
                              void* d_out, int out_size, void* d_ws, size_t ws_size,
                              hipStream_t stream) {
  const float* emb    = (const float*)d_in[0]; // [B,DIM,T]
  const float* W_in   = (const float*)d_in[1]; // [Q,DIM,CD]
  const float* b_in   = (const float*)d_in[2]; // [Q,CD]
  const float* W_out  = (const float*)d_in[3]; // [Q,CD,DIM]
  const float* b_out  = (const float*)d_in[4]; // [Q,DIM]
  const float* embedt = (const float*)d_in[5]; // [Q,K,CD]

  float* R = (float*)d_ws;                         // residual, 64 MB
  size_t relem = (size_t)B_ * DIM_ * T_;
  int*   idx = (int*)(R + relem);                  // [M]
  float* enormAdj = (float*)(idx + M_);            // [Q,K]

  hipMemcpyAsync(R, emb, relem * sizeof(float), hipMemcpyDeviceToDevice, stream);
  rvq_enorm<<<(Q_ * K_ + 255) / 256, 256, 0, stream>>>(embedt, b_in, enormAdj);

  for (int q = 0; q < Q_; ++q) {
    const float* Wi = W_in   + (size_t)q * DIM_ * CD_;
    const float* E  = embedt + (size_t)q * K_ * CD_;
    const float* Wo = W_out  + (size_t)q * CD_ * DIM_;
    const float* bo = b_out  + (size_t)q * DIM_;
    rvq_assign<<<M_ / 64, 128, 0, stream>>>(R, Wi, E, enormAdj + (size_t)q * K_, idx);
    rvq_decode<<<M_ / 64, 128, 0, stream>>>(R, E, Wo, bo, idx);
  }
  rvq_final<<<4096, 256, 0, stream>>>((const float4*)emb, (const float4*)R,
                                      (float4*)d_out, (int)(relem / 4));
}